// ChessGNN_62251255988645
// MI455X (gfx1250) — compile-verified
//
#include <hip/hip_runtime.h>
#include <hip/hip_bf16.h>
#include <math.h>

#define NN 200000
#define EE 1200000
#define GG 2048
#define HH 64
#define NEG_SLOPE 0.2f

typedef float v2f __attribute__((ext_vector_type(2)));
typedef float v8f __attribute__((ext_vector_type(8)));

// ---------------- helpers ----------------

__device__ __forceinline__ void atomicMaxF(float* addr, float val) {
    // order-preserving int/uint trick; memory must be initialized to -inf
    if (!signbit(val)) {
        atomicMax((int*)addr, __float_as_int(val));
    } else {
        atomicMin((unsigned int*)addr, __float_as_uint(val));
    }
}

// CDNA5 async global->LDS copy (16B per lane), tracked by ASYNCcnt.
// VDST operand = LDS byte offset (low 32 bits of a generic LDS pointer).
__device__ __forceinline__ void async_copy_b128(unsigned lds_off, const void* gaddr) {
    asm volatile("global_load_async_to_lds_b128 %0, %1, off"
                 :: "v"(lds_off), "v"(gaddr)
                 : "memory");
}
__device__ __forceinline__ void async_wait0() {
    asm volatile("s_wait_asynccnt 0" ::: "memory");
}

// ---------------- embed: h = x @ W(5x64) + b ----------------

__global__ void gnn_embed(const float* __restrict__ x, const float* __restrict__ W,
                          const float* __restrict__ b, float* __restrict__ h) {
    __shared__ float sW[5 * 64];
    __shared__ float sb[64];
    for (int i = threadIdx.x; i < 5 * 64; i += blockDim.x) sW[i] = W[i];
    if (threadIdx.x < 64) sb[threadIdx.x] = b[threadIdx.x];
    __syncthreads();
    int t = blockIdx.x * blockDim.x + threadIdx.x;
    if (t >= NN * HH) return;
    int n = t >> 6, j = t & 63;
    float acc = sb[j];
#pragma unroll
    for (int k = 0; k < 5; ++k) acc += x[n * 5 + k] * sW[k * 64 + j];
    h[t] = acc;
}

// ---------------- WMMA GEMM: Hout[N,64] = Hin[N,64] @ W[64,64] ----------------
// One block = 256 threads = 8 waves, covering a 32-row x 64-col output tile.
// Wave w: column tile (w&3), row sub-tile (w>>2). fp32 exact via V_WMMA_F32_16X16X4_F32.
// W (16 KB) and the 32-row A tile (8 KB) are staged into LDS with async
// global->LDS b128 copies (no VGPR round-trip), synced via s_wait_asynccnt.

__global__ void gnn_wmma_gemm64(const float* __restrict__ Hin,
                                const float* __restrict__ W,
                                float* __restrict__ Hout) {
    __shared__ float sW[64 * 64];   // 16 KB
    __shared__ float sA[32 * 64];   // 8 KB
    const int row0 = blockIdx.x * 32;

    const unsigned sW_off = (unsigned)(size_t)&sW[0];
    const unsigned sA_off = (unsigned)(size_t)&sA[0];
    // W: 1024 x 16B chunks; A tile: 512 x 16B chunks
#pragma unroll
    for (int p = 0; p < 4; ++p) {
        int i = threadIdx.x + p * 256;
        async_copy_b128(sW_off + i * 16, W + i * 4);
    }
#pragma unroll
    for (int p = 0; p < 2; ++p) {
        int i = threadIdx.x + p * 256;
        async_copy_b128(sA_off + i * 16, Hin + (size_t)row0 * 64 + i * 4);
    }
    async_wait0();
    __syncthreads();

    const int wave = threadIdx.x >> 5;
    const int lane = threadIdx.x & 31;
    const int ct   = wave & 3;       // column tile 0..3
    const int rt   = wave >> 2;      // row tile 0..1
    const int col0 = ct * 16;
    const int r    = lane & 15;      // M (for A) / N (for B) within tile
    const int hi   = lane >> 4;      // half-wave selector

    v8f c = {};
    const float* Arow = &sA[(rt * 16 + r) * 64];
#pragma unroll
    for (int k0 = 0; k0 < 64; k0 += 4) {
        const int ka = k0 + hi * 2;
        v2f a, bfrag;
        a.x = Arow[ka];
        a.y = Arow[ka + 1];
        bfrag.x = sW[ka * 64 + col0 + r];
        bfrag.y = sW[(ka + 1) * 64 + col0 + r];
        c = __builtin_amdgcn_wmma_f32_16x16x4_f32(
                /*neg_a=*/false, a, /*neg_b=*/false, bfrag,
                /*c_mod=*/(short)0, c, /*reuse_a=*/false, /*reuse_b=*/false);
    }

    // C/D layout: VGPR v -> M = v (lanes 0-15) or v+8 (lanes 16-31), N = lane&15
    const int col   = col0 + r;
    const int rbase = row0 + rt * 16 + hi * 8;
#pragma unroll
    for (int v = 0; v < 8; ++v) {
        Hout[(size_t)(rbase + v) * 64 + col] = c[v];
    }
}

// ---------------- attention coefficients: alpha_s/d[n] = h[n] . a_src/dst ----------------
// one wave32 per node; each lane covers features (lane, lane+32)

__global__ void gnn_attn_coeff(const float* __restrict__ h,
                               const float* __restrict__ asrc, const float* __restrict__ adst,
                               float* __restrict__ als, float* __restrict__ ald) {
    int wid  = (blockIdx.x * blockDim.x + threadIdx.x) >> 5;
    int lane = threadIdx.x & 31;
    if (wid >= NN) return;
    const float* row = h + (size_t)wid * 64;
    float v0 = row[lane], v1 = row[lane + 32];
    float s = v0 * asrc[lane] + v1 * asrc[lane + 32];
    float d = v0 * adst[lane] + v1 * adst[lane + 32];
#pragma unroll
    for (int off = 16; off > 0; off >>= 1) {
        s += __shfl_xor(s, off, 32);
        d += __shfl_xor(d, off, 32);
    }
    if (lane == 0) { als[wid] = s; ald[wid] = d; }
}

// ---------------- per-layer init: out=0, m=-inf, denom=0 ----------------

__global__ void gnn_layer_init(float* __restrict__ out, float* __restrict__ m,
                               float* __restrict__ denom) {
    int t = blockIdx.x * blockDim.x + threadIdx.x;
    if (t >= NN * HH) return;
    out[t] = 0.0f;
    if (t < NN) { m[t] = -INFINITY; denom[t] = 0.0f; }
}

// ---------------- edge pass 1: e = leaky_relu(as[src]+ad[dst]); segment max ----------------

__global__ void gnn_edge_max(const long long* __restrict__ ei,
                             const float* __restrict__ als, const float* __restrict__ ald,
                             float* __restrict__ ebuf, float* __restrict__ m) {
    int t = blockIdx.x * blockDim.x + threadIdx.x;
    if (t >= EE + NN) return;
    int s, d;
    if (t < EE) { s = (int)ei[t]; d = (int)ei[EE + t]; } else { s = d = t - EE; }
    float e = als[s] + ald[d];
    e = (e > 0.0f) ? e : NEG_SLOPE * e;
    ebuf[t] = e;
    atomicMaxF(&m[d], e);
}

// ---------------- edge pass 2: w = exp(e - m[dst]); segment sum ----------------

__global__ void gnn_edge_expsum(const long long* __restrict__ ei,
                                const float* __restrict__ m,
                                float* __restrict__ ebuf, float* __restrict__ denom) {
    int t = blockIdx.x * blockDim.x + threadIdx.x;
    if (t >= EE + NN) return;
    int d = (t < EE) ? (int)ei[EE + t] : (t - EE);
    float w = __expf(ebuf[t] - m[d]);
    ebuf[t] = w;
    atomicAdd(&denom[d], w);
}

// ---------------- edge pass 3: out[dst] += h[src] * (w / denom[dst]) ----------------
// 64 threads per edge (feature-parallel, coalesced gather & atomic scatter)

__global__ void gnn_edge_aggr(const long long* __restrict__ ei,
                              const float* __restrict__ h, const float* __restrict__ ebuf,
                              const float* __restrict__ denom, float* __restrict__ out) {
    long long t = (long long)blockIdx.x * blockDim.x + threadIdx.x;
    if (t >= (long long)(EE + NN) * 64) return;
    int eidx = (int)(t >> 6);
    int j    = (int)(t & 63);
    int s, d;
    if (eidx < EE) { s = (int)ei[eidx]; d = (int)ei[EE + eidx]; } else { s = d = eidx - EE; }
    float alpha = ebuf[eidx] / (denom[d] + 1e-16f);
    atomicAdd(&out[(size_t)d * 64 + j], h[(size_t)s * 64 + j] * alpha);
}

// ---------------- bias (+ optional relu) ----------------

__global__ void gnn_bias_act(float* __restrict__ h, const float* __restrict__ b, int do_relu) {
    int t = blockIdx.x * blockDim.x + threadIdx.x;
    if (t >= NN * HH) return;
    float v = h[t] + b[t & 63];
    h[t] = do_relu ? fmaxf(v, 0.0f) : v;
}

// ---------------- pooling ----------------

__global__ void gnn_pool_init(float* __restrict__ psum, float* __restrict__ pmax,
                              float* __restrict__ cnt) {
    int t = blockIdx.x * blockDim.x + threadIdx.x;
    if (t >= GG * HH) return;
    psum[t] = 0.0f;
    pmax[t] = -INFINITY;
    if (t < GG) cnt[t] = 0.0f;
}

__global__ void gnn_pool_scatter(const float* __restrict__ h, const long long* __restrict__ batch,
                                 float* __restrict__ psum, float* __restrict__ pmax,
                                 float* __restrict__ cnt) {
    int t = blockIdx.x * blockDim.x + threadIdx.x;
    if (t >= NN * HH) return;
    int n = t >> 6, j = t & 63;
    int g = (int)batch[n];
    float v = h[t];
    atomicAdd(&psum[g * 64 + j], v);
    atomicMaxF(&pmax[g * 64 + j], v);
    if (j == 0) atomicAdd(&cnt[g], 1.0f);
}

__global__ void gnn_pool_finalize(const float* __restrict__ psum, const float* __restrict__ pmax,
                                  const float* __restrict__ cnt, float* __restrict__ gfeat) {
    int t = blockIdx.x * blockDim.x + threadIdx.x;
    if (t >= GG * 2 * HH) return;
    int g = t >> 7, j = t & 127;
    float c = cnt[g];
    if (j < 64) {
        gfeat[t] = psum[g * 64 + j] / fmaxf(c, 1.0f);
    } else {
        gfeat[t] = (c > 0.0f) ? pmax[g * 64 + (j - 64)] : 0.0f;
    }
}

// ---------------- readout MLP ----------------

__global__ void gnn_fc1(const float* __restrict__ gfeat, const float* __restrict__ W,
                        const float* __restrict__ b, float* __restrict__ o1) {
    int t = blockIdx.x * blockDim.x + threadIdx.x;
    if (t >= GG * 64) return;
    int g = t >> 6, j = t & 63;
    float acc = b[j];
    const float* row = gfeat + (size_t)g * 128;
#pragma unroll 4
    for (int k = 0; k < 128; ++k) acc += row[k] * W[k * 64 + j];
    o1[t] = fmaxf(acc, 0.0f);
}

__global__ void gnn_fc2(const float* __restrict__ o1, const float* __restrict__ W,
                        const float* __restrict__ b, float* __restrict__ o2) {
    int t = blockIdx.x * blockDim.x + threadIdx.x;
    if (t >= GG * 32) return;
    int g = t >> 5, j = t & 31;
    float acc = b[j];
    const float* row = o1 + (size_t)g * 64;
#pragma unroll 4
    for (int k = 0; k < 64; ++k) acc += row[k] * W[k * 32 + j];
    o2[t] = fmaxf(acc, 0.0f);
}

__global__ void gnn_fc3(const float* __restrict__ o2, const float* __restrict__ W,
                        const float* __restrict__ b, float* __restrict__ out) {
    int g = blockIdx.x * blockDim.x + threadIdx.x;
    if (g >= GG) return;
    float acc = b[0];
    const float* row = o2 + (size_t)g * 32;
#pragma unroll
    for (int k = 0; k < 32; ++k) acc += row[k] * W[k];
    out[g] = acc;
}

// ---------------- launcher ----------------

extern "C" void kernel_launch(void* const* d_in, const int* in_sizes, int n_in,
                              void* d_out, int out_size, void* d_ws, size_t ws_size,
                              hipStream_t stream) {
    const float*     x       = (const float*)d_in[0];
    const long long* ei      = (const long long*)d_in[1];
    const long long* batch   = (const long long*)d_in[2];
    const float*     embed_W = (const float*)d_in[3];
    const float*     embed_b = (const float*)d_in[4];
    const float*     g1_W    = (const float*)d_in[5];
    const float*     g1_as   = (const float*)d_in[6];
    const float*     g1_ad   = (const float*)d_in[7];
    const float*     g1_b    = (const float*)d_in[8];
    const float*     g2_W    = (const float*)d_in[9];
    const float*     g2_as   = (const float*)d_in[10];
    const float*     g2_ad   = (const float*)d_in[11];
    const float*     g2_b    = (const float*)d_in[12];
    const float*     fc1_W   = (const float*)d_in[13];
    const float*     fc1_b   = (const float*)d_in[14];
    const float*     fc2_W   = (const float*)d_in[15];
    const float*     fc2_b   = (const float*)d_in[16];
    const float*     fc3_W   = (const float*)d_in[17];
    const float*     fc3_b   = (const float*)d_in[18];
    float*           out     = (float*)d_out;

    // workspace layout
    char* base = (char*)d_ws;
    size_t off = 0;
    auto alloc = [&](size_t nelem) { float* p = (float*)(base + off); off += nelem * sizeof(float); return p; };
    float* h_a   = alloc((size_t)NN * HH);     // ping buffer (also aggregation output)
    float* h_b   = alloc((size_t)NN * HH);     // pong buffer (h' = h @ W)
    float* als   = alloc(NN);
    float* ald   = alloc(NN);
    float* m     = alloc(NN);
    float* denom = alloc(NN);
    float* ebuf  = alloc((size_t)(EE + NN));
    float* psum  = alloc((size_t)GG * HH);
    float* pmax  = alloc((size_t)GG * HH);
    float* cnt   = alloc(GG);
    float* gfeat = alloc((size_t)GG * 2 * HH);
    float* o1    = alloc((size_t)GG * 64);
    float* o2    = alloc((size_t)GG * 32);
    (void)ws_size; (void)n_in; (void)in_sizes; (void)out_size;

    const int B = 256;
    const int gNH   = (NN * HH + B - 1) / B;           // 50000
    const int gWMMA = NN / 32;                          // 6250 (N divisible by 32)
    const int gATTN = (NN + 7) / 8;                     // 8 nodes (waves) per block
    const int gEDGE = (EE + NN + B - 1) / B;            // 5469
    const int gAGGR = (int)(((long long)(EE + NN) * 64 + B - 1) / B); // 350000
    const int gPOOLI = (GG * HH + B - 1) / B;
    const int gFIN  = (GG * 2 * HH + B - 1) / B;

    // embed
    gnn_embed<<<gNH, B, 0, stream>>>(x, embed_W, embed_b, h_a);

    // ---- GAT layer 1 ----
    gnn_wmma_gemm64<<<gWMMA, B, 0, stream>>>(h_a, g1_W, h_b);
    gnn_attn_coeff<<<gATTN, B, 0, stream>>>(h_b, g1_as, g1_ad, als, ald);
    gnn_layer_init<<<gNH, B, 0, stream>>>(h_a, m, denom);
    gnn_edge_max<<<gEDGE, B, 0, stream>>>(ei, als, ald, ebuf, m);
    gnn_edge_expsum<<<gEDGE, B, 0, stream>>>(ei, m, ebuf, denom);
    gnn_edge_aggr<<<gAGGR, B, 0, stream>>>(ei, h_b, ebuf, denom, h_a);
    gnn_bias_act<<<gNH, B, 0, stream>>>(h_a, g1_b, /*relu=*/1);

    // ---- GAT layer 2 ----
    gnn_wmma_gemm64<<<gWMMA, B, 0, stream>>>(h_a, g2_W, h_b);
    gnn_attn_coeff<<<gATTN, B, 0, stream>>>(h_b, g2_as, g2_ad, als, ald);
    gnn_layer_init<<<gNH, B, 0, stream>>>(h_a, m, denom);
    gnn_edge_max<<<gEDGE, B, 0, stream>>>(ei, als, ald, ebuf, m);
    gnn_edge_expsum<<<gEDGE, B, 0, stream>>>(ei, m, ebuf, denom);
    gnn_edge_aggr<<<gAGGR, B, 0, stream>>>(ei, h_b, ebuf, denom, h_a);
    gnn_bias_act<<<gNH, B, 0, stream>>>(h_a, g2_b, /*relu=*/0);

    // ---- pooling ----
    gnn_pool_init<<<gPOOLI, B, 0, stream>>>(psum, pmax, cnt);
    gnn_pool_scatter<<<gNH, B, 0, stream>>>(h_a, batch, psum, pmax, cnt);
    gnn_pool_finalize<<<gFIN, B, 0, stream>>>(psum, pmax, cnt, gfeat);

    // ---- MLP readout ----
    gnn_fc1<<<(GG * 64 + B - 1) / B, B, 0, stream>>>(gfeat, fc1_W, fc1_b, o1);
    gnn_fc2<<<(GG * 32 + B - 1) / B, B, 0, stream>>>(o1, fc2_W, fc2_b, o2);
    gnn_fc3<<<(GG + B - 1) / B, B, 0, stream>>>(o2, fc3_W, fc3_b, out);
}